// RandomWalk_21646635172577
// MI455X (gfx1250) — compile-verified
//
#include <hip/hip_runtime.h>

// out[b, p, c] = x[b, S-1, c]  (random-walk forecast; param*0 == 0)
// Pure store-bandwidth problem: ~318 MB written, ~441 KB read, 0 FLOPs.
// Roofline on MI455X: 318MB / 23.3 TB/s ~= 13.6 us. No matrix math -> no WMMA;
// the CDNA5 paths that matter are async global->LDS staging and NT b128 stores.

typedef float v4f __attribute__((ext_vector_type(4)));

static constexpr int B  = 128;
static constexpr int S  = 512;
static constexpr int P  = 720;
static constexpr int C  = 862;        // 3448 B per row: 8B-aligned, NOT 16B-periodic
static constexpr int DP = 2 * C;      // double pattern: 1724 floats = 6896 B = 431 x 16B
static constexpr int CH = DP / 4;     // 431 float4 chunks per double-row
static constexpr int DR = P / 2;      // 360 double-rows per batch
static constexpr int BLOCKS_PER_BATCH = 8;   // 360 / 8 = 45 double-rows per block
static constexpr int THREADS = 448;          // 14 wave32; 431 lanes active in store loop

__global__ __launch_bounds__(THREADS) void RandomWalk_bcast_kernel(
    const float* __restrict__ x, float* __restrict__ out) {
  __shared__ __align__(16) float pat[DP];

  const int t = threadIdx.x;
  const int b = blockIdx.y;

  // Last observed row for this batch: x[b, S-1, :]  (862 floats, 8B-aligned)
  const float* __restrict__ last = x + ((size_t)b * S + (S - 1)) * C;

  // ---- Stage first copy of the row into LDS with async global->LDS b64 ----
  // 862 floats == exactly 431 x 8-byte chunks; one chunk per lane (t < 431).
  if (t < C / 2) {
    unsigned lds_off =
        (unsigned)(size_t)(&pat[0]) + (unsigned)(t * 8);   // low 32 bits = LDS byte addr
    unsigned long long gaddr =
        (unsigned long long)(size_t)last + (unsigned long long)(t * 8);
    asm volatile("global_load_async_to_lds_b64 %0, %1, off"
                 :: "v"(lds_off), "v"(gaddr)
                 : "memory");
  }
  asm volatile("s_wait_asynccnt 0" ::: "memory");
  __syncthreads();

  // ---- Second copy so 2 rows (6896 B) form a 16B-periodic pattern ----
  for (int i = t; i < C; i += THREADS) pat[C + i] = pat[i];
  __syncthreads();

  // Each active lane owns one 16B chunk of the double-row; it is loop-invariant,
  // so hoist it into registers (ds_load_b128 once) and keep the hot loop as
  // pure non-temporal global_store_b128.
  v4f v = {};
  if (t < CH) v = ((const v4f*)pat)[t];

  v4f* __restrict__ out4 = (v4f*)(out + (size_t)b * P * C);  // batch base is 16B-aligned
  if (t < CH) {
    for (int r = blockIdx.x; r < DR; r += BLOCKS_PER_BATCH) {
      __builtin_nontemporal_store(v, &out4[(size_t)r * CH + t]);
    }
  }
}

extern "C" void kernel_launch(void* const* d_in, const int* in_sizes, int n_in,
                              void* d_out, int out_size, void* d_ws, size_t ws_size,
                              hipStream_t stream) {
  (void)in_sizes; (void)n_in; (void)out_size; (void)d_ws; (void)ws_size;
  const float* x = (const float*)d_in[0];   // [B, S, C] float32
  float* out = (float*)d_out;               // [B, P, C] float32

  dim3 grid(BLOCKS_PER_BATCH, B);
  RandomWalk_bcast_kernel<<<grid, THREADS, 0, stream>>>(x, out);
}